// Model_54941221651154
// MI455X (gfx1250) — compile-verified
//
#include <hip/hip_runtime.h>

// Problem dims (fixed by the reference).
#define BB 2
#define TT 1024
#define HH 8
#define DD 64
#define NLEVC 11
#define NWAVE 4   // waves per workgroup: column tiles round-robined over waves

typedef __attribute__((ext_vector_type(2))) float v2f;
typedef __attribute__((ext_vector_type(8))) float v8f;

// ---------------------------------------------------------------------------
// Kernel 1: inclusive cumsum of gates along T for each (b,h).
// g: [B,T,H] -> cs: [B*H, T] in workspace.  A[i,j] = exp(cs[i]-cs[j]).
// ---------------------------------------------------------------------------
__global__ __launch_bounds__(TT) void cumsum_g_kernel(const float* __restrict__ g,
                                                      float* __restrict__ cs) {
  const int bh = blockIdx.x;
  const int b = bh / HH, h = bh % HH;
  __shared__ float buf[TT];
  const int t = threadIdx.x;
  buf[t] = g[(b * TT + t) * HH + h];
  __syncthreads();
#pragma unroll
  for (int off = 1; off < TT; off <<= 1) {
    float prev = (t >= off) ? buf[t - off] : 0.0f;
    __syncthreads();
    if (t >= off) buf[t] += prev;
    __syncthreads();
  }
  cs[bh * TT + t] = buf[t];
}

// ---------------------------------------------------------------------------
// Kernel 2: hierarchical-decay attention.
// One workgroup (4 wave32) per 16-row query tile of one (b,h):
//   wave w handles column tiles ct = w, w+NWAVE, ...  (triangular loop)
//     scores = q k^T          (fp32 WMMA 16x16x4, K=64 as 4 independent
//                              4-deep chains -> 4-way ILP in the matrix pipe)
//     Hm     = exp(cs_i-cs_j) * level_scales[b,h,lev(i,j),i]  (elementwise)
//     oacc  += Hm @ v         (fp32 WMMA 16x16x4, K=16, 4 independent N-tiles)
//   then the 4 partial 16x64 tiles are reduced through LDS and each wave
//   stores one 16-wide slab of the output.
// lev(i,j) = 0 on the diagonal, else 1 + floor(log2(i^j)) (closed form of the
// reference's hierarchical band masks: the highest differing bit m satisfies
// (i>>m)-(j>>m)==1 with bit m of i set, which is exactly the band condition).
// ---------------------------------------------------------------------------
__global__ __launch_bounds__(32 * NWAVE) void hattn_kernel(
    const float* __restrict__ q, const float* __restrict__ k,
    const float* __restrict__ v, const float* __restrict__ ls,
    const float* __restrict__ cs, float* __restrict__ out) {
  const int rt   = blockIdx.x;          // query row tile (0..63)
  const int bh   = blockIdx.y;          // b*H + h        (0..15)
  const int b    = bh / HH;
  const int h    = bh % HH;
  const int w    = threadIdx.x >> 5;    // wave id in workgroup
  const int lane = threadIdx.x & 31;    // wave32 lane
  const int n    = lane & 15;           // N index / M index for A operands
  const int hi   = lane >> 4;           // K-half selector
  const int row0 = rt * 16;

  __shared__ float mtile[NWAVE * 16 * 16];   // per-wave C->A relayout bounce
  __shared__ float red[NWAVE * 16 * 64];     // partial output reduction
  float* mt = mtile + w * 256;

  // --- q tile in A-operand layout: lane holds row M=n, K = 4*kc + 2*hi + r.
  // Consecutive K pairs are contiguous in memory (D innermost) -> v2f loads.
  v2f qa[16];
  {
    const float* qrow = q + (((size_t)(b * TT + row0 + n)) * HH + h) * DD;
#pragma unroll
    for (int kc = 0; kc < 16; ++kc)
      qa[kc] = *(const v2f*)(qrow + 4 * kc + 2 * hi);
  }

  // --- per-lane row cumsum values (fixed for the whole block).
  const float* csbh = cs + (size_t)bh * TT;
  float csi[8];
#pragma unroll
  for (int r = 0; r < 8; ++r) csi[r] = csbh[row0 + r + 8 * hi];

  v8f oacc[4];                          // partial out tile 16x64 (4 x 16x16 C)
#pragma unroll
  for (int ng = 0; ng < 4; ++ng) oacc[ng] = (v8f)(0.0f);

  for (int ct = w; ct <= rt; ct += NWAVE) {  // triangular, strided over waves
    const int col0 = ct * 16;

    // k tile in B-operand layout (same lane map: N=n, K=4*kc+2*hi+r).
    const float* krow = k + (((size_t)(b * TT + col0 + n)) * HH + h) * DD;
    if (ct + NWAVE <= rt)  // prefetch this wave's next k row -> global_prefetch_b8
      __builtin_prefetch(krow + NWAVE * 16 * HH * DD, 0, 3);
    v2f kb[16];
#pragma unroll
    for (int kc = 0; kc < 16; ++kc)
      kb[kc] = *(const v2f*)(krow + 4 * kc + 2 * hi);

    // scores[l,c] = sum_d q[l,d]*k[c,d], D=64.  Four independent 4-deep WMMA
    // chains (K-chunks strided by 4) avoid the 16-deep RAW accumulator chain.
    v8f sp[4];
#pragma unroll
    for (int u = 0; u < 4; ++u) sp[u] = (v8f)(0.0f);
#pragma unroll
    for (int kc = 0; kc < 16; ++kc)
      sp[kc & 3] = __builtin_amdgcn_wmma_f32_16x16x4_f32(
          false, qa[kc], false, kb[kc], (short)0, sp[kc & 3], false, false);
    v8f sacc = (sp[0] + sp[1]) + (sp[2] + sp[3]);

    // Elementwise hierarchical decay. C layout: elem (i=row0+r+8*hi, j=col0+n).
    const int   j   = col0 + n;
    const float csj = csbh[j];
#pragma unroll
    for (int r = 0; r < 8; ++r) {
      const int i = row0 + r + 8 * hi;
      float wgt = 0.0f;
      if (j <= i) {
        const int lev = (i == j) ? 0
                                 : (32 - __builtin_clz(((unsigned)(i ^ j)) | 1u));
        const float scale = ls[((size_t)bh * NLEVC + lev) * TT + i];
        wgt = __expf(csi[r] - csj) * scale;   // v_exp_f32
      }
      sacc[r] *= wgt;
    }

    // Re-layout M: C layout -> A layout via this wave's LDS region (in-order).
#pragma unroll
    for (int r = 0; r < 8; ++r)
      mt[(r + 8 * hi) * 16 + n] = sacc[r];
    __builtin_amdgcn_wave_barrier();
    v2f ma[4];
#pragma unroll
    for (int kc = 0; kc < 4; ++kc)
      ma[kc] = *(const v2f*)(&mt[n * 16 + 4 * kc + 2 * hi]);
    __builtin_amdgcn_wave_barrier();

    // oacc[16x64] += M[16x16] @ v[16x64]: 4 N-groups x 4 K-chunks of WMMA
    // (independent accumulators per N-group -> 4-way ILP).
    const float* vbase = v + (((size_t)(b * TT + col0)) * HH + h) * DD;
#pragma unroll
    for (int ng = 0; ng < 4; ++ng) {
#pragma unroll
      for (int kc = 0; kc < 4; ++kc) {
        v2f vb;  // B layout: N = n (contig. across lanes), K = 4*kc+2*hi+r
        vb.x = vbase[(size_t)(4 * kc + 2 * hi + 0) * (HH * DD) + 16 * ng + n];
        vb.y = vbase[(size_t)(4 * kc + 2 * hi + 1) * (HH * DD) + 16 * ng + n];
        oacc[ng] = __builtin_amdgcn_wmma_f32_16x16x4_f32(
            false, ma[kc], false, vb, (short)0, oacc[ng], false, false);
      }
    }
  }

  // --- cross-wave reduction of the 4 partial tiles through LDS.
#pragma unroll
  for (int ng = 0; ng < 4; ++ng)
#pragma unroll
    for (int r = 0; r < 8; ++r)
      red[w * (16 * 64) + (r + 8 * hi) * 64 + 16 * ng + n] = oacc[ng][r];
  __syncthreads();

  // Wave w sums and stores output columns [16w, 16w+16): out[b, l, h, p].
#pragma unroll
  for (int r = 0; r < 8; ++r) {
    const int m = r + 8 * hi;              // row within tile
    const int p = 16 * w + n;              // output feature
    float acc = 0.0f;
#pragma unroll
    for (int pw = 0; pw < NWAVE; ++pw)
      acc += red[pw * (16 * 64) + m * 64 + p];
    out[(((size_t)(b * TT + row0 + m)) * HH + h) * DD + p] = acc;
  }
}

// ---------------------------------------------------------------------------
extern "C" void kernel_launch(void* const* d_in, const int* in_sizes, int n_in,
                              void* d_out, int out_size, void* d_ws, size_t ws_size,
                              hipStream_t stream) {
  const float* q  = (const float*)d_in[0];   // [B,T,H,D]
  const float* k  = (const float*)d_in[1];   // [B,T,H,D]
  const float* v  = (const float*)d_in[2];   // [B,T,H,D]
  const float* g  = (const float*)d_in[3];   // [B,T,H]
  const float* ls = (const float*)d_in[4];   // [B,H,NLEV,T]
  float* out = (float*)d_out;                // [B,T,H,D]
  float* cs  = (float*)d_ws;                 // [B*H, T] gate cumsums (64 KB)

  cumsum_g_kernel<<<dim3(BB * HH), dim3(TT), 0, stream>>>(g, cs);
  hattn_kernel<<<dim3(TT / 16, BB * HH), dim3(32 * NWAVE), 0, stream>>>(
      q, k, v, ls, cs, out);
}